// NodePairGaussianKernel_30288109371690
// MI455X (gfx1250) — compile-verified
//
#include <hip/hip_runtime.h>

typedef __attribute__((ext_vector_type(2))) float v2f;
typedef __attribute__((ext_vector_type(4))) float v4f;
typedef __attribute__((ext_vector_type(8))) float v8f;

#define NN 512            // nodes per (batch, bin)
#define FF 16             // feature dim
#define LDS_STRIDE 18     // padded row stride (floats): keeps float2 loads 8B aligned
#define EPSV 1e-6f
// -DIST_MULT * log2(e): exp(-0.1*d) == exp2(NEG_DM_LOG2E * d)
#define NEG_DM_LOG2E (-0.14426950408889634f)

__launch_bounds__(256, 2)
__global__ void node_pair_gaussian_kernel(const float* __restrict__ x,
                                          const float* __restrict__ msk,
                                          float* __restrict__ out) {
    __shared__ float Alds[NN * LDS_STRIDE];   // masked A panel, 512 x 16 (padded)
    __shared__ float na[NN];                  // row squared norms

    const int bb  = blockIdx.x;               // batch*bin slice, 0..511
    const int tid = threadIdx.x;              // 0..255

    const float* xb = x   + (size_t)bb * NN * FF;
    const float* mb = msk + (size_t)bb * NN;
    float*       ob = out + (size_t)bb * NN * NN;

    // ---- Stage A = x * msk into LDS, compute row norms (2 rows per thread) ----
    for (int r = tid; r < NN; r += 256) {
        const float m = mb[r];
        const v4f* src = (const v4f*)(xb + r * FF);
        float* dst = &Alds[r * LDS_STRIDE];
        float acc = 0.f;
#pragma unroll
        for (int q = 0; q < 4; ++q) {
            v4f v = src[q];
            v.x *= m; v.y *= m; v.z *= m; v.w *= m;
            acc += v.x * v.x + v.y * v.y + v.z * v.z + v.w * v.w;
            dst[q * 4 + 0] = v.x;
            dst[q * 4 + 1] = v.y;
            dst[q * 4 + 2] = v.z;
            dst[q * 4 + 3] = v.w;
        }
        na[r] = acc;
    }
    __syncthreads();

    const int wave = tid >> 5;      // 8 wave32 waves / workgroup
    const int lane = tid & 31;
    const int half = lane >> 4;     // lane group (0: lanes 0-15, 1: lanes 16-31)
    const int lm   = lane & 15;

    // ---- 32x32 grid of 16x16 output tiles; adjacent waves take adjacent j-tiles ----
    for (int t = wave; t < 1024; t += 8) {
        const int ti = t >> 5;
        const int tj = t & 31;

        // Fragment source rows in the A panel (B = A^T -> identical addressing)
        const int ri = ti * 16 + lm;
        const int rj = tj * 16 + lm;

        v8f c = {};
#pragma unroll
        for (int s = 0; s < 4; ++s) {
            // f32 16x16x4 layout: lane L, VGPR j -> K = 4s + 2*(L/16) + j
            const int col = s * 4 + half * 2;
            v2f a = *(const v2f*)&Alds[ri * LDS_STRIDE + col];
            v2f b = *(const v2f*)&Alds[rj * LDS_STRIDE + col];
            // D = A*B + C ; chain C through all 4 K-steps (K = 16 total)
            c = __builtin_amdgcn_wmma_f32_16x16x4_f32(
                    /*neg_a=*/false, a, /*neg_b=*/false, b,
                    /*c_mod=*/(short)0, c, /*reuse_a=*/false, /*reuse_b=*/false);
        }

        // Epilogue: out = exp2(-0.1*log2e * sqrt(max(na_i - 2G + na_j, eps)))
        // Inputs to sqrt are >= 1e-6 (far above the 2^-96 denorm-rescue range),
        // so the raw v_sqrt_f32 is safe; exp2 arg is strictly negative so the
        // result is already in (0,1) and the reference clip is a no-op.
        const float ncol = na[tj * 16 + lm];
        const int   row0 = ti * 16 + half * 8;   // C/D layout: M = r + 8*half
        const int   col  = tj * 16 + lm;         // N = lane % 16
#pragma unroll
        for (int r = 0; r < 8; ++r) {
            const int row = row0 + r;
            float d2 = fmaxf(na[row] - 2.0f * c[r] + ncol, EPSV);
            float d  = __builtin_amdgcn_sqrtf(d2);       // raw v_sqrt_f32 (TRANS)
            float e  = __builtin_amdgcn_exp2f(d * NEG_DM_LOG2E); // v_exp_f32 (TRANS)
            ob[(size_t)row * NN + col] = e;
        }
    }
}

extern "C" void kernel_launch(void* const* d_in, const int* in_sizes, int n_in,
                              void* d_out, int out_size, void* d_ws, size_t ws_size,
                              hipStream_t stream) {
    (void)in_sizes; (void)n_in; (void)d_ws; (void)ws_size; (void)out_size;
    const float* x   = (const float*)d_in[0];  // (8, 64, 512, 16) f32
    const float* msk = (const float*)d_in[1];  // (8, 64, 512, 1) f32
    float* out = (float*)d_out;                // (8, 64, 512, 512, 1) f32

    dim3 grid(8 * 64);   // one workgroup per (batch, bin)
    dim3 block(256);     // 8 wave32 waves
    node_pair_gaussian_kernel<<<grid, block, 0, stream>>>(x, msk, out);
}